// PVConvLayer_5471788335253
// MI455X (gfx1250) — compile-verified
//
#include <hip/hip_runtime.h>

typedef __attribute__((ext_vector_type(2))) float v2f;
typedef __attribute__((ext_vector_type(8))) float v8f;

#define VXD 160
#define VYD 160
#define VZD 32
#define CCH 64
#define NPTS 200000
#define BN_EPS 1e-3f
#define NPAD 68   // 64 data floats + 4 pad -> stride 68 kills LDS bank conflicts

// ---------------------------------------------------------------------------
// Prep: transpose conv weights [tap][cin][cout] -> [tap][cout][cin] and
// W_mlp [cin][cout] -> [cout][cin] so WMMA B-fragments are contiguous b64 loads
// ---------------------------------------------------------------------------
__global__ void prep_weights(const float* __restrict__ conv_w,
                             const float* __restrict__ w_mlp,
                             float* __restrict__ wt_conv,
                             float* __restrict__ wt_mlp) {
    int gid = blockIdx.x * blockDim.x + threadIdx.x;
    if (gid < 27 * 64 * 64) {
        int tap = gid >> 12;
        int r = gid & 4095;
        int cout = r >> 6;
        int cin = r & 63;
        wt_conv[gid] = conv_w[(tap << 12) + (cin << 6) + cout];
    } else {
        int g = gid - 27 * 64 * 64;
        if (g < 64 * 64) {
            int cout = g >> 6;
            int cin = g & 63;
            wt_mlp[g] = w_mlp[(cin << 6) + cout];
        }
    }
}

// ---------------------------------------------------------------------------
// Voxelize: scatter-add raw point features into the grid
// ---------------------------------------------------------------------------
__global__ void scatter_add(const float* __restrict__ inputs,
                            const int* __restrict__ vidx,
                            float* __restrict__ grid_in) {
    int gid = blockIdx.x * blockDim.x + threadIdx.x;
    if (gid >= NPTS * CCH) return;
    int n = gid >> 6;
    int c = gid & 63;
    int ix = vidx[3 * n + 0];
    int iy = vidx[3 * n + 1];
    int iz = vidx[3 * n + 2];
    atomicAdd(&grid_in[((size_t)(ix * VYD + iy) * VZD + iz) * CCH + c], inputs[gid]);
}

// scatter-mean: scale each voxel by 1/count
__global__ void scale_grid(float* __restrict__ grid_in,
                           const float* __restrict__ inv) {
    int gid = blockIdx.x * blockDim.x + threadIdx.x;
    if (gid >= VXD * VYD * VZD * 16) return;
    int v = gid >> 4;
    int c4 = gid & 15;
    float s = inv[v];
    float4* p = (float4*)(grid_in + (size_t)v * CCH + c4 * 4);
    float4 val = *p;
    val.x *= s; val.y *= s; val.z *= s; val.w *= s;
    *p = val;
}

// ---------------------------------------------------------------------------
// Point MLP: [N,64] @ [64,64] + bias, ReLU, BN  — fp32 WMMA 16x16x4
// Block = 128 threads = 4 waves; each wave owns 16 couts; block owns 16 points.
// ---------------------------------------------------------------------------
__global__ __launch_bounds__(128) void mlp_wmma(
    const float* __restrict__ inputs, const float* __restrict__ wt_mlp,
    const float* __restrict__ b_mlp,
    const float* __restrict__ gamma, const float* __restrict__ beta,
    const float* __restrict__ mean, const float* __restrict__ var,
    float* __restrict__ out) {
    int tid = threadIdx.x;
    int wave = tid >> 5;
    int lane = tid & 31;
    int m = lane & 15;       // A row (point) / B col (cout) per WMMA layout
    int hi = lane >> 4;      // 0: K=k0,k0+1   1: K=k0+2,k0+3
    int cout = wave * 16 + m;
    int n0 = blockIdx.x * 16;

    float sc = gamma[cout] * rsqrtf(var[cout] + BN_EPS);
    float sh = beta[cout] - mean[cout] * sc;
    float bias = b_mlp[cout];

    const float* arow = inputs + (size_t)(n0 + m) * 64 + 2 * hi;
    const float* brow = wt_mlp + cout * 64 + 2 * hi;

    v8f acc = {};
#pragma unroll
    for (int k0 = 0; k0 < 64; k0 += 4) {
        v2f a = *(const v2f*)(arow + k0);
        v2f b = *(const v2f*)(brow + k0);
        acc = __builtin_amdgcn_wmma_f32_16x16x4_f32(
            false, a, false, b, (short)0, acc, false, false);
    }
#pragma unroll
    for (int j = 0; j < 8; ++j) {
        int M = j + 8 * hi;  // C/D layout: VGPR j = rows j (lanes 0-15), j+8 (lanes 16-31)
        float v = fmaxf(acc[j] + bias, 0.0f) * sc + sh;
        out[(size_t)(n0 + M) * 64 + cout] = v;
    }
}

// ---------------------------------------------------------------------------
// Conv3D 3x3x3 SAME as implicit GEMM — fp32 WMMA 16x16x4, dual-tile.
// Workgroup = (x, y-pair, 16-voxel z strip). Each wave holds TWO 16x16
// accumulators (y0 and y1 tiles) so one weight B-fragment feeds 2 WMMAs.
// LDS slab: 3(dx) x 4(dy span) x 18(z) x 64ch, padded stride 68 -> 58.7 KB.
// ---------------------------------------------------------------------------
__global__ __launch_bounds__(128) void conv_wmma(
    const float* __restrict__ grid_in, const float* __restrict__ wt,
    const float* __restrict__ conv_b,
    const float* __restrict__ gamma, const float* __restrict__ beta,
    const float* __restrict__ mean, const float* __restrict__ var,
    float* __restrict__ grid_out) {
    __shared__ __align__(16) float slab[12 * 18 * NPAD];  // 14688 floats

    int bid = blockIdx.x;          // 160 * 80 * 2
    int zt = bid & 1;
    int r = bid >> 1;
    int x = r / (VYD / 2);
    int y0 = (r % (VYD / 2)) * 2;  // tiles at y0 and y0+1
    int zbase = zt * 16;
    int tid = threadIdx.x;

    // Stage padded slab: 12 (dx,dyi) columns x 18 z x 64 ch, zero OOB.
    // Column sc = dxi*4 + dyi covers x2 = x+dxi-1, y2 = y0+dyi-1.
    for (int idx = tid; idx < 12 * 18 * 16; idx += 128) {
        int c4 = idx & 15;
        int t = idx >> 4;
        int zl = t % 18;
        int sc = t / 18;
        int x2 = x + (sc >> 2) - 1;
        int y2 = y0 + (sc & 3) - 1;
        int z2 = zbase + zl - 1;
        float4 v = make_float4(0.f, 0.f, 0.f, 0.f);
        if ((unsigned)x2 < (unsigned)VXD && (unsigned)y2 < (unsigned)VYD &&
            (unsigned)z2 < (unsigned)VZD)
            v = *(const float4*)(grid_in +
                                 ((size_t)(x2 * VYD + y2) * VZD + z2) * CCH + c4 * 4);
        *(float4*)(&slab[(sc * 18 + zl) * NPAD + c4 * 4]) = v;
    }
    __syncthreads();

    int wave = tid >> 5;
    int lane = tid & 31;
    int m = lane & 15;
    int hi = lane >> 4;
    int cout = wave * 16 + m;

    float sc2 = gamma[cout] * rsqrtf(var[cout] + BN_EPS);
    float sh2 = beta[cout] - mean[cout] * sc2;
    float bias = conv_b[cout];

    v8f acc0 = {};
    v8f acc1 = {};
    for (int tap = 0; tap < 27; ++tap) {
        int kd = tap / 9;                  // dx = kd-1
        int kh = (tap / 3) % 3;            // dy = kh-1
        int dz = tap % 3 - 1;              // kw-1
        // tile0 (y = y0):   dyi = kh;  tile1 (y = y0+1): dyi = kh+1
        const float* a0row =
            &slab[((kd * 4 + kh) * 18 + (m + dz + 1)) * NPAD + 2 * hi];
        const float* a1row = a0row + 18 * NPAD;
        const float* brow = wt + tap * 4096 + cout * 64 + 2 * hi;
#pragma unroll
        for (int k0 = 0; k0 < 64; k0 += 4) {
            v2f b = *(const v2f*)(brow + k0);    // global b64, shared by 2 WMMAs
            v2f a0 = *(const v2f*)(a0row + k0);  // ds_load_b64, conflict-free
            v2f a1 = *(const v2f*)(a1row + k0);
            acc0 = __builtin_amdgcn_wmma_f32_16x16x4_f32(
                false, a0, false, b, (short)0, acc0, false, false);
            acc1 = __builtin_amdgcn_wmma_f32_16x16x4_f32(
                false, a1, false, b, (short)0, acc1, false, false);
        }
    }

    size_t ob0 = (size_t)(x * VYD + y0) * VZD + zbase;
    size_t ob1 = ob0 + VZD;  // y0+1
#pragma unroll
    for (int j = 0; j < 8; ++j) {
        int M = j + 8 * hi;
        float v0 = fmaxf(acc0[j] + bias, 0.0f) * sc2 + sh2;
        float v1 = fmaxf(acc1[j] + bias, 0.0f) * sc2 + sh2;
        grid_out[(ob0 + M) * CCH + cout] = v0;
        grid_out[(ob1 + M) * CCH + cout] = v1;
    }
}

// ---------------------------------------------------------------------------
// Devoxelize: trilinear gather from conv output, accumulate into d_out
// ---------------------------------------------------------------------------
__global__ void trilinear_add(const float* __restrict__ grid_out,
                              const float* __restrict__ pts,
                              float* __restrict__ out) {
    int gid = blockIdx.x * blockDim.x + threadIdx.x;
    if (gid >= NPTS * 16) return;
    int n = gid >> 4;
    int c4 = gid & 15;

    float px = fminf(fmaxf(pts[3 * n + 0], 0.f), (float)(VXD - 1));
    float py = fminf(fmaxf(pts[3 * n + 1], 0.f), (float)(VYD - 1));
    float pz = fminf(fmaxf(pts[3 * n + 2], 0.f), (float)(VZD - 1));
    float fx = floorf(px), fy = floorf(py), fz = floorf(pz);
    int ix0 = (int)fx, iy0 = (int)fy, iz0 = (int)fz;
    int ix1 = min(ix0 + 1, VXD - 1);
    int iy1 = min(iy0 + 1, VYD - 1);
    int iz1 = min(iz0 + 1, VZD - 1);
    float tx = px - fx, ty = py - fy, tz = pz - fz;

    int xs[2] = {ix0, ix1}; float wx[2] = {1.f - tx, tx};
    int ys[2] = {iy0, iy1}; float wy[2] = {1.f - ty, ty};
    int zs[2] = {iz0, iz1}; float wz[2] = {1.f - tz, tz};

    float4 rsum = make_float4(0.f, 0.f, 0.f, 0.f);
#pragma unroll
    for (int cx = 0; cx < 2; ++cx)
#pragma unroll
        for (int cy = 0; cy < 2; ++cy)
#pragma unroll
            for (int cz = 0; cz < 2; ++cz) {
                float w = wx[cx] * wy[cy] * wz[cz];
                const float4 g = *(const float4*)(
                    grid_out + ((size_t)(xs[cx] * VYD + ys[cy]) * VZD + zs[cz]) * CCH + c4 * 4);
                rsum.x += g.x * w; rsum.y += g.y * w;
                rsum.z += g.z * w; rsum.w += g.w * w;
            }

    float4* o = (float4*)(out + (size_t)n * CCH + c4 * 4);
    float4 cur = *o;
    cur.x += rsum.x; cur.y += rsum.y; cur.z += rsum.z; cur.w += rsum.w;
    *o = cur;
}

// ---------------------------------------------------------------------------
extern "C" void kernel_launch(void* const* d_in, const int* in_sizes, int n_in,
                              void* d_out, int out_size, void* d_ws, size_t ws_size,
                              hipStream_t stream) {
    (void)in_sizes; (void)n_in; (void)out_size; (void)ws_size;
    const float* inputs  = (const float*)d_in[0];
    const float* pt_crd  = (const float*)d_in[1];
    const float* ppv_inv = (const float*)d_in[2];
    const float* W_mlp   = (const float*)d_in[3];
    const float* b_mlp   = (const float*)d_in[4];
    const float* gamma1  = (const float*)d_in[5];
    const float* beta1   = (const float*)d_in[6];
    const float* mean1   = (const float*)d_in[7];
    const float* var1    = (const float*)d_in[8];
    const float* conv_w  = (const float*)d_in[9];
    const float* conv_b  = (const float*)d_in[10];
    const float* gamma2  = (const float*)d_in[11];
    const float* beta2   = (const float*)d_in[12];
    const float* mean2   = (const float*)d_in[13];
    const float* var2    = (const float*)d_in[14];
    const int*   vidx    = (const int*)d_in[15];
    float* out = (float*)d_out;

    const size_t GRID_ELEMS = (size_t)VXD * VYD * VZD * CCH;  // 52,428,800
    float* ws       = (float*)d_ws;
    float* grid_in  = ws;
    float* grid_out = ws + GRID_ELEMS;
    float* wt_conv  = ws + 2 * GRID_ELEMS;
    float* wt_mlp   = wt_conv + 27 * 64 * 64;

    // zero scatter target each call (replay-deterministic)
    hipMemsetAsync(grid_in, 0, GRID_ELEMS * sizeof(float), stream);

    prep_weights<<<(27 * 64 * 64 + 64 * 64 + 255) / 256, 256, 0, stream>>>(
        conv_w, W_mlp, wt_conv, wt_mlp);
    scatter_add<<<(NPTS * CCH + 255) / 256, 256, 0, stream>>>(inputs, vidx, grid_in);
    scale_grid<<<(VXD * VYD * VZD * 16 + 255) / 256, 256, 0, stream>>>(grid_in, ppv_inv);
    mlp_wmma<<<NPTS / 16, 128, 0, stream>>>(
        inputs, wt_mlp, b_mlp, gamma1, beta1, mean1, var1, out);
    conv_wmma<<<VXD * (VYD / 2) * 2, 128, 0, stream>>>(
        grid_in, wt_conv, conv_b, gamma2, beta2, mean2, var2, grid_out);
    trilinear_add<<<(NPTS * 16 + 255) / 256, 256, 0, stream>>>(grid_out, pt_crd, out);
}